// Head_46643344834719
// MI455X (gfx1250) — compile-verified
//
#include <hip/hip_runtime.h>

typedef __attribute__((ext_vector_type(2))) float v2f;
typedef __attribute__((ext_vector_type(8))) float v8f;

#define DIM 512
#define BATCH 512

// ---------------------------------------------------------------------------
// Phase 1: fused q/k/v projection GEMMs via V_WMMA_F32_16X16X4_F32.
// C[m, n] = sum_k x[m, k] * W[n, k] + bias[n]   (torch Linear: x @ W^T + b)
// One wave32 per 16(M) x 32(N) output tile: one A (x) load feeds two WMMAs
// into two independent accumulator chains (breaks RAW-on-C serialization and
// halves x-tile traffic). blockIdx.z selects which of q/k/v.
//
// f32 WMMA operand layout (ISA 7.12.2):
//   A (16x4, MxK):  lanes 0-15: M=lane,    VGPR0=K0, VGPR1=K1
//                   lanes 16-31: M=lane-16, VGPR0=K2, VGPR1=K3
//   B (4x16, KxN):  mirrored: lane holds column N=lane%16, K pair by lane half
//   C/D (16x16):    VGPR r: M = r + 8*(lane/16), N = lane%16
// Since B[k,n] = W[n,k], both A and B are contiguous float2 row-major loads.
// ---------------------------------------------------------------------------
__global__ __launch_bounds__(32) void qkv_wmma_kernel(
    const float* __restrict__ x,
    const float* __restrict__ Wq, const float* __restrict__ bq,
    const float* __restrict__ Wk, const float* __restrict__ bk,
    const float* __restrict__ Wv, const float* __restrict__ bv,
    float* __restrict__ q, float* __restrict__ k, float* __restrict__ v)
{
    const int o0    = blockIdx.x * 32;   // output-feature tile (N), 2 sub-tiles
    const int m0    = blockIdx.y * 16;   // batch tile (M)
    const int which = blockIdx.z;        // 0=q, 1=k, 2=v

    const float* __restrict__ W    = (which == 0) ? Wq : (which == 1) ? Wk : Wv;
    const float* __restrict__ bias = (which == 0) ? bq : (which == 1) ? bk : bv;
    float* __restrict__ out        = (which == 0) ? q  : (which == 1) ? k  : v;

    const int lane = threadIdx.x;
    const int half = lane >> 4;   // 0: K pair {0,1}, 1: K pair {2,3}
    const int l16  = lane & 15;

    const float* __restrict__ arow  = x + (size_t)(m0 + l16) * DIM + 2 * half;
    const float* __restrict__ brow0 = W + (size_t)(o0 + l16) * DIM + 2 * half;
    const float* __restrict__ brow1 = W + (size_t)(o0 + 16 + l16) * DIM + 2 * half;

    v8f c0 = {};
    v8f c1 = {};
#pragma unroll 4
    for (int kk = 0; kk < DIM; kk += 4) {
        v2f a  = *(const v2f*)(arow + kk);
        v2f b0 = *(const v2f*)(brow0 + kk);
        v2f b1 = *(const v2f*)(brow1 + kk);
        c0 = __builtin_amdgcn_wmma_f32_16x16x4_f32(
                 false, a, false, b0, (short)0, c0, false, false);
        c1 = __builtin_amdgcn_wmma_f32_16x16x4_f32(
                 false, a, false, b1, (short)0, c1, false, false);
    }

    const float bcol0 = bias[o0 + l16];
    const float bcol1 = bias[o0 + 16 + l16];
#pragma unroll
    for (int r = 0; r < 8; ++r) {
        const size_t row = (size_t)(m0 + r + 8 * half) * DIM;
        out[row + (o0 + l16)]      = c0[r] + bcol0;
        out[row + (o0 + 16 + l16)] = c1[r] + bcol1;
    }
}

// ---------------------------------------------------------------------------
// Phase 2: rank-1 softmax attention. One block (512 threads) per batch b.
// out[b,i] = sum_j exp(s_i*k_j - m_i) * v_j / sum_j exp(s_i*k_j - m_i)
// with s_i = q[b,i]/sqrt(d), m_i = max_j s_i*k_j = s_i>=0 ? s_i*kmax : s_i*kmin
// (reproduces jax.nn.softmax's max subtraction exactly).
//
// TRANS-pipe bound: 512^3 v_exp_f32 total. (k,v) pairs are interleaved in LDS
// so one ds_load_b128 feeds two exp iterations (k and v for two j's): 0.5 DS
// ops per exp instead of 2. Two independent num/den accumulator pairs keep
// the VALU FMA chains off the exp critical path.
// ---------------------------------------------------------------------------
__global__ __launch_bounds__(DIM) void attn_kernel(
    const float* __restrict__ q, const float* __restrict__ k,
    const float* __restrict__ v, float* __restrict__ out)
{
    __shared__ __align__(16) float2 kvs[DIM];   // (k_j, v_j) interleaved, 4 KB
    __shared__ float redmax[DIM / 32];
    __shared__ float redmin[DIM / 32];

    const int b = blockIdx.x;
    const int i = threadIdx.x;

    const float kval = k[(size_t)b * DIM + i];
    const float vval = v[(size_t)b * DIM + i];
    kvs[i] = make_float2(kval, vval);

    // wave32 min/max reduction, then cross-wave via LDS
    float mx = kval, mn = kval;
#pragma unroll
    for (int off = 16; off > 0; off >>= 1) {
        mx = fmaxf(mx, __shfl_xor(mx, off, 32));
        mn = fminf(mn, __shfl_xor(mn, off, 32));
    }
    if ((i & 31) == 0) { redmax[i >> 5] = mx; redmin[i >> 5] = mn; }
    __syncthreads();

    float kmax = redmax[0], kmin = redmin[0];
#pragma unroll
    for (int w = 1; w < DIM / 32; ++w) {
        kmax = fmaxf(kmax, redmax[w]);
        kmin = fminf(kmin, redmin[w]);
    }

    const float scale = 0.04419417382415922f;  // 1/sqrt(512)
    const float s = q[(size_t)b * DIM + i] * scale;
    const float m = (s >= 0.f) ? s * kmax : s * kmin;

    float num0 = 0.f, den0 = 0.f;
    float num1 = 0.f, den1 = 0.f;
#pragma unroll 4
    for (int j = 0; j < DIM; j += 2) {
        const float4 p = *(const float4*)&kvs[j];   // ds_load_b128: k0,v0,k1,v1
        const float e0 = __expf(fmaf(s, p.x, -m));  // v_exp_f32 (TRANS)
        const float e1 = __expf(fmaf(s, p.z, -m));
        num0 = fmaf(e0, p.y, num0);
        den0 += e0;
        num1 = fmaf(e1, p.w, num1);
        den1 += e1;
    }
    out[(size_t)b * DIM + i] = (num0 + num1) / (den0 + den1);
}

// ---------------------------------------------------------------------------
extern "C" void kernel_launch(void* const* d_in, const int* in_sizes, int n_in,
                              void* d_out, int out_size, void* d_ws, size_t ws_size,
                              hipStream_t stream) {
    (void)in_sizes; (void)n_in; (void)out_size; (void)ws_size;

    const float* x  = (const float*)d_in[0];
    const float* Wq = (const float*)d_in[1];
    const float* bq = (const float*)d_in[2];
    const float* Wk = (const float*)d_in[3];
    const float* bk = (const float*)d_in[4];
    const float* Wv = (const float*)d_in[5];
    const float* bv = (const float*)d_in[6];
    float* out = (float*)d_out;

    float* q = (float*)d_ws;                 // 1 MB
    float* k = q + (size_t)BATCH * DIM;      // 1 MB
    float* v = k + (size_t)BATCH * DIM;      // 1 MB

    dim3 grid1(DIM / 32, BATCH / 16, 3);     // 16 x 32 tiles x {q,k,v}
    qkv_wmma_kernel<<<grid1, 32, 0, stream>>>(x, Wq, bq, Wk, bk, Wv, bv, q, k, v);

    attn_kernel<<<BATCH, DIM, 0, stream>>>(q, k, v, out);
}